// SoftDotBlockAttention_1434519077498
// MI455X (gfx1250) — compile-verified
//
#include <hip/hip_runtime.h>
#include <hip/hip_bf16.h>
#include <math.h>

#define B_  64
#define S_  2048
#define D_  512
#define NB_ 8

typedef __attribute__((ext_vector_type(2))) float v2f;
typedef __attribute__((ext_vector_type(8))) float v8f;

// ---------------------------------------------------------------------------
// Kernel 1: target = h @ W^T using V_WMMA_F32_16X16X4_F32 (full f32 precision)
//   target[m][n] = sum_k h[m][k] * W[n][k]
// One wave (32 lanes) per 16x16 output tile, K looped in steps of 4.
// A-matrix (16x4, MxK): lane = M (both halves), VGPR0 = K {0|2}, VGPR1 = K {1|3}
//   -> per-lane contiguous float2 from h row.
// B-matrix (4x16, KxN): lane = N, same K pairing -> per-lane contiguous float2
//   along row n of W (since B[k][n] = W[n][k]).
// C/D (16x16 f32): VGPR i = row (i + 8*half), col = lane&15.
// ---------------------------------------------------------------------------
__global__ __launch_bounds__(32)
void target_gemm_wmma(const float* __restrict__ h,
                      const float* __restrict__ W,
                      float* __restrict__ target) {
    const int lane = threadIdx.x & 31;
    const int half = lane >> 4;       // 0: lanes 0-15, 1: lanes 16-31
    const int lm   = lane & 15;
    const int m0   = blockIdx.y << 4; // 4 tiles over B=64
    const int n0   = blockIdx.x << 4; // 32 tiles over D=512

    const float* arow = h + (size_t)(m0 + lm) * D_;
    const float* brow = W + (size_t)(n0 + lm) * D_;   // B[k][n] = W[n][k]

    v8f acc = {};
#pragma unroll 8
    for (int k0 = 0; k0 < D_; k0 += 4) {
        const int kb = k0 + 2 * half;
        v2f a = *(const v2f*)(arow + kb);
        v2f b = *(const v2f*)(brow + kb);
        // 8 args: (neg_a, A, neg_b, B, c_mod, C, reuse_a, reuse_b)
        acc = __builtin_amdgcn_wmma_f32_16x16x4_f32(
            false, a, false, b, (short)0, acc, false, false);
    }
#pragma unroll
    for (int i = 0; i < 8; ++i)
        target[(size_t)(m0 + i + 8 * half) * D_ + (n0 + lm)] = acc[i];
}

// ---------------------------------------------------------------------------
// Kernel 2: attn[b,s] = context[b,s,:] . target[b,:]
// Bandwidth-bound: streams 268 MB of context. One block per (b, 64-row chunk),
// 8 waves; each wave handles 8 consecutive rows. target[b,:] staged in LDS.
// Per row: 4 x b128 loads per lane (full 2 KB row across the wave), FMAs,
// then a 5-step wave32 shuffle reduction.
// ---------------------------------------------------------------------------
__global__ __launch_bounds__(256)
void attn_kernel(const float* __restrict__ context,
                 const float* __restrict__ target,
                 float* __restrict__ attn) {
    __shared__ __align__(16) float t_lds[D_];
    const int b   = blockIdx.y;
    const int tid = threadIdx.x;

    // stage target[b,:] (512 floats) into LDS: 256 threads x float2
    ((float2*)t_lds)[tid] = ((const float2*)(target + (size_t)b * D_))[tid];
    __syncthreads();

    const int wave   = tid >> 5;
    const int lane   = tid & 31;
    const int s_base = blockIdx.x * 64 + wave * 8;

    for (int r = 0; r < 8; ++r) {
        const int s = s_base + r;
        const float* row = context + ((size_t)b * S_ + s) * D_;
        // hint the next row into cache (speculative; dropped if invalid)
        __builtin_prefetch(row + D_, 0, 0);

        float partial = 0.f;
#pragma unroll
        for (int i = 0; i < 4; ++i) {
            const int d = i * 128 + lane * 4;
            const float4 c = *(const float4*)(row + d);
            const float4 t = *(const float4*)(t_lds + d);
            partial = fmaf(c.x, t.x, partial);
            partial = fmaf(c.y, t.y, partial);
            partial = fmaf(c.z, t.z, partial);
            partial = fmaf(c.w, t.w, partial);
        }
#pragma unroll
        for (int off = 16; off > 0; off >>= 1)
            partial += __shfl_xor(partial, off, 32);
        if (lane == 0) attn[(size_t)b * S_ + s] = partial;
    }
}

// ---------------------------------------------------------------------------
// Kernel 3: per-batch masked softmax over the selected block (L in [1,64))
// and weighted_context[b,:] = sum_{j<L} p[j] * context[b, sc+j, :].
// One block per batch. Wave 0 does the softmax (<=63 elems) with shuffle
// reductions; then 256 threads accumulate 2 d-columns each over the block.
// ---------------------------------------------------------------------------
__global__ __launch_bounds__(256)
void block_softmax_wc(const float* __restrict__ attn,
                      const float* __restrict__ context,
                      const int*   __restrict__ lengths,
                      const int*   __restrict__ sel,
                      float*       __restrict__ wc) {
    const int b   = blockIdx.x;
    const int tid = threadIdx.x;
    __shared__ float p[64];
    __shared__ float s_inv;
    __shared__ int   s_sc, s_L;

    if (tid == 0) {
        const int idx = sel[b];
        int start = 0;
        for (int i = 0; i < idx; ++i) start += lengths[b * NB_ + i];
        s_sc = start + 1;
        s_L  = lengths[b * NB_ + idx];
    }
    __syncthreads();
    const int sc = s_sc;
    const int L  = s_L;           // 1..63

    if (tid < 32) {
        const int lane = tid;
        const float NEG = -3.0e38f;
        const float* arow = attn + (size_t)b * S_ + sc;
        float v0 = (lane      < L) ? arow[lane]      : NEG;
        float v1 = (lane + 32 < L) ? arow[lane + 32] : NEG;
        float m = fmaxf(v0, v1);
#pragma unroll
        for (int off = 16; off > 0; off >>= 1)
            m = fmaxf(m, __shfl_xor(m, off, 32));
        float e0 = (lane      < L) ? __expf(v0 - m) : 0.f;
        float e1 = (lane + 32 < L) ? __expf(v1 - m) : 0.f;
        float ssum = e0 + e1;
#pragma unroll
        for (int off = 16; off > 0; off >>= 1)
            ssum += __shfl_xor(ssum, off, 32);
        if (lane      < L) p[lane]      = e0;
        if (lane + 32 < L) p[lane + 32] = e1;
        if (lane == 0) s_inv = 1.0f / ssum;
    }
    __syncthreads();

    const float inv = s_inv;
    const int d0 = tid;
    const int d1 = tid + 256;
    float acc0 = 0.f, acc1 = 0.f;
    const float* base = context + ((size_t)b * S_ + sc) * D_;
    for (int j = 0; j < L; ++j) {
        const float pj = p[j] * inv;
        acc0 = fmaf(pj, base[(size_t)j * D_ + d0], acc0);
        acc1 = fmaf(pj, base[(size_t)j * D_ + d1], acc1);
    }
    wc[(size_t)b * D_ + d0] = acc0;
    wc[(size_t)b * D_ + d1] = acc1;
}

// ---------------------------------------------------------------------------
extern "C" void kernel_launch(void* const* d_in, const int* in_sizes, int n_in,
                              void* d_out, int out_size, void* d_ws, size_t ws_size,
                              hipStream_t stream) {
    (void)in_sizes; (void)n_in; (void)out_size; (void)ws_size;

    const float* h        = (const float*)d_in[0];   // [B, D]
    const float* context  = (const float*)d_in[1];   // [B, S, D]
    const float* W        = (const float*)d_in[2];   // [D, D]
    const int*   lengths  = (const int*)  d_in[3];   // [B, NB]
    const int*   sel      = (const int*)  d_in[4];   // [B]

    float* out_wc   = (float*)d_out;                  // [B, D]
    float* out_attn = out_wc + (size_t)B_ * D_;       // [B, S]
    float* target   = (float*)d_ws;                   // [B, D] scratch (128 KB)

    // 1) target = h @ W^T  (WMMA f32 16x16x4): 32 N-tiles x 4 M-tiles
    target_gemm_wmma<<<dim3(D_ / 16, B_ / 16), 32, 0, stream>>>(h, W, target);

    // 2) attn = einsum('bsd,bd->bs'): stream 268 MB of context
    attn_kernel<<<dim3(S_ / 64, B_), 256, 0, stream>>>(context, target, out_attn);

    // 3) selected-block softmax + weighted context
    block_softmax_wc<<<B_, 256, 0, stream>>>(out_attn, context, lengths, sel, out_wc);
}